// Sampler_49830210568854
// MI455X (gfx1250) — compile-verified
//
#include <hip/hip_runtime.h>
#include <hip/hip_bf16.h>
#include <stdint.h>

#define THREADS        256
#define TDM_TILE       4000          // floats per TDM tile (tile_dim0, fits 16-bit)
#define TDM_TILE_BYTES 16000
#define ASY_TILE       1024          // floats per async-pipeline tile (pass B)
#define ASY_TILE_BYTES 4096
#define NBINS          4096
#define CAP            1024
#define TOPM           64
#define SAMPLING_EPS   1e-5f

typedef uint32_t u32x4 __attribute__((ext_vector_type(4)));
typedef uint32_t u32x8 __attribute__((ext_vector_type(8)));

// Monotonic float -> uint mapping (order preserving for all finite floats)
__device__ __forceinline__ uint32_t f2ord(float f) {
    uint32_t x = __float_as_uint(f);
    return x ^ ((x & 0x80000000u) ? 0xFFFFFFFFu : 0x80000000u);
}

// ---- CDNA5 Tensor Data Mover: 1-D tile, global -> LDS, tracked by TENSORcnt.
// D# group0: count=1, lds_addr, 57-bit global_addr, type=2 ("image").
// D# group1: data_size=4B, tensor_dim0 = tile_dim0 = nelem (standalone 1-D tensor).
__device__ __forceinline__ void tdm_load_1d(uint64_t gaddr, uint32_t lds_addr, uint32_t nelem) {
    u32x4 g0;
    g0[0] = 1u;                                                    // count=1, user mode
    g0[1] = lds_addr;                                              // lds_addr [63:32]
    g0[2] = (uint32_t)gaddr;                                       // global_addr lo
    g0[3] = ((uint32_t)(gaddr >> 32) & 0x01FFFFFFu) | 0x80000000u; // addr[56:32] | type=2
    u32x8 g1;
    g1[0] = 0x00020000u;                       // workgroup_mask=0, data_size=2 (4 bytes)
    g1[1] = (nelem & 0xFFFFu) << 16;           // tensor_dim0[15:0] @ bits 63:48
    g1[2] = ((nelem >> 16) & 0xFFFFu) | (1u << 16); // tensor_dim0[31:16] | tensor_dim1=1
    g1[3] = (nelem & 0xFFFFu) << 16;           // tensor_dim1[31:16]=0 | tile_dim0
    g1[4] = 0u;                                // tile_dim1=0 (1-D), tile_dim2=0
    g1[5] = nelem;                             // tensor_dim0_stride lo32
    g1[6] = 0u;                                // stride hi | tensor_dim1_stride lo
    g1[7] = 0u;
    asm volatile("tensor_load_to_lds %0, %1" :: "s"(g0), "s"(g1) : "memory");
}

// ---- CDNA5 async copy: 16 bytes per lane, global -> LDS, tracked by ASYNCcnt.
// GVS addressing: 64-bit SGPR base + 32-bit VGPR offset.
__device__ __forceinline__ void async_cp16(uint32_t lds_addr, uint32_t goff, uint64_t base) {
    asm volatile("global_load_async_to_lds_b128 %0, %1, %2"
                 :: "v"(lds_addr), "v"(goff), "s"(base)
                 : "memory");
}
__device__ __forceinline__ void wait_async_le1() {
    asm volatile("s_wait_asynccnt 0x1" ::: "memory");
}
__device__ __forceinline__ void wait_async_le0() {
    asm volatile("s_wait_asynccnt 0x0" ::: "memory");
}

__global__ __launch_bounds__(THREADS) void sampler_kernel(
    const float* __restrict__ logits,
    const float* __restrict__ temperature,
    const int*   __restrict__ top_k,
    const float* __restrict__ top_p,
    const float* __restrict__ noise,
    float* __restrict__ out,
    int V, int B)
{
    __shared__ __align__(16) uint8_t s_raw[2][TDM_TILE_BYTES];  // TDM / async tile buffers
    __shared__ unsigned s_hist[NBINS];
    __shared__ float    s_cval[CAP];
    __shared__ int      s_cidx[CAP];
    __shared__ float    s_tval[TOPM];
    __shared__ int      s_tidx[TOPM];
    __shared__ float    s_e[TOPM];
    __shared__ float    s_cs[TOPM];
    __shared__ int      s_keep[TOPM];
    __shared__ unsigned s_ccount;
    __shared__ unsigned s_uth;

    const int tid = threadIdx.x;
    const int row = blockIdx.x;
    const float* rowptr = logits + (size_t)row * V;
    const uint64_t base = (uint64_t)(uintptr_t)rowptr;
    const float temp  = temperature[row];
    const float scale = (temp < SAMPLING_EPS) ? 1.0f : (1.0f / temp);
    const float NINF  = -__builtin_inff();

    for (int b = tid; b < NBINS; b += THREADS) s_hist[b] = 0u;
    if (tid == 0) s_ccount = 0u;
    __syncthreads();

    const uint32_t lds_raw = (uint32_t)(uintptr_t)(&s_raw[0][0]);

    // ============ Pass A: histogram; bulk streaming via TDM (TENSORcnt) ============
    {
        const int ntt = V / TDM_TILE;
        if (tid < 32 && ntt > 0)
            tdm_load_1d(base, lds_raw, TDM_TILE);                       // tile 0 -> buf0
        for (int t = 0; t < ntt; ++t) {
            const int cur = t & 1;
            const int nxt = cur ^ 1;
            if (tid < 32) {
                if (t + 1 < ntt) {
                    tdm_load_1d(base + (uint64_t)(t + 1) * TDM_TILE_BYTES,
                                lds_raw + (uint32_t)nxt * TDM_TILE_BYTES, TDM_TILE);
                    __builtin_amdgcn_s_wait_tensorcnt(1);               // current tile done
                } else {
                    __builtin_amdgcn_s_wait_tensorcnt(0);
                }
            }
            __syncthreads();                                            // release consumers
            const float* tp = (const float*)s_raw[cur];
            for (int i = tid; i < TDM_TILE; i += THREADS) {
                float f = tp[i] * scale;
                atomicAdd(&s_hist[f2ord(f) >> 20], 1u);
            }
            __syncthreads();                                            // buffer reuse safe
        }
        for (int i = ntt * TDM_TILE + tid; i < V; i += THREADS) {       // tail
            float f = rowptr[i] * scale;
            atomicAdd(&s_hist[f2ord(f) >> 20], 1u);
        }
        __syncthreads();
    }

    // threshold = lower bound of the bin containing the 64th largest element
    if (tid == 0) {
        unsigned acc = 0; int tb = 0;
        for (int b = NBINS - 1; b >= 0; --b) {
            acc += s_hist[b];
            if (acc >= TOPM) { tb = b; break; }
        }
        s_uth = ((uint32_t)tb) << 20;
    }
    __syncthreads();
    const uint32_t uth = s_uth;

    // ===== Pass B: collect candidates >= threshold; per-lane async copies (ASYNCcnt) =====
    {
        const int nta = V / ASY_TILE;
        if (nta > 0)
            async_cp16(lds_raw + (uint32_t)(tid * 16), (uint32_t)(tid * 16), base);
        for (int t = 0; t < nta; ++t) {
            const int cur = t & 1;
            const int nxt = cur ^ 1;
            if (t + 1 < nta) {
                async_cp16(lds_raw + (uint32_t)(nxt * TDM_TILE_BYTES + tid * 16),
                           (uint32_t)((t + 1) * ASY_TILE_BYTES + tid * 16), base);
                wait_async_le1();
            } else {
                wait_async_le0();
            }
            __syncthreads();
            const float* bp = (const float*)s_raw[cur];
            #pragma unroll
            for (int e = 0; e < 4; ++e) {
                float f = bp[tid * 4 + e] * scale;
                if (f2ord(f) >= uth) {
                    unsigned slot = atomicAdd(&s_ccount, 1u);
                    if (slot < CAP) { s_cval[slot] = f; s_cidx[slot] = t * ASY_TILE + tid * 4 + e; }
                }
            }
            __syncthreads();
        }
        for (int i = nta * ASY_TILE + tid; i < V; i += THREADS) {
            float f = rowptr[i] * scale;
            if (f2ord(f) >= uth) {
                unsigned slot = atomicAdd(&s_ccount, 1u);
                if (slot < CAP) { s_cval[slot] = f; s_cidx[slot] = i; }
            }
        }
        __syncthreads();
    }

    // ===================== Finalize on wave 0 (wave32) =====================
    int C = (int)s_ccount; if (C > CAP) C = CAP;
    if (tid < 32) {
        const int lane = tid;
        volatile float* cv = s_cval;
        volatile int*   ci = s_cidx;
        const int m = (C < TOPM) ? C : TOPM;

        // repeated wave-parallel argmax -> descending top-m (tie: lower index)
        for (int j = 0; j < m; ++j) {
            float bv = NINF; int bi = 0x7fffffff; int bs = 0;
            for (int t2 = lane; t2 < C; t2 += 32) {
                float v = cv[t2]; int id = ci[t2];
                if (v > bv || (v == bv && id < bi)) { bv = v; bi = id; bs = t2; }
            }
            for (int off = 16; off > 0; off >>= 1) {
                float ov = __shfl_xor(bv, off, 32);
                int   oi = __shfl_xor(bi, off, 32);
                int   os = __shfl_xor(bs, off, 32);
                if (ov > bv || (ov == bv && oi < bi)) { bv = ov; bi = oi; bs = os; }
            }
            if (lane == 0) { s_tval[j] = bv; s_tidx[j] = bi; cv[bs] = NINF; }
        }

        volatile float* ve    = s_e;
        volatile int*   vkeep = s_keep;
        volatile float* vtv   = s_tval;
        volatile int*   vti   = s_tidx;

        int sc = 0; float Z2 = 0.f;
        if (lane == 0 && m > 0) {
            const float vmax = vtv[0];
            int kk = top_k[row]; if (kk < 1) kk = 1; if (kk > m) kk = m;
            const float pivot = vtv[kk - 1];           // k-th largest; v < pivot -> -inf
            const float p = top_p[row];
            float Z1 = 0.f;
            for (int j = 0; j < m; ++j) {
                float vv = vtv[j];
                float e = (vv >= pivot) ? __expf(vv - vmax) : 0.f;
                ve[j] = e; Z1 += e;
            }
            float suf = 0.f;                            // ascending cumsum == suffix sum
            for (int j = m - 1; j >= 0; --j) { suf += ve[j]; s_cs[j] = suf; }
            const float thr = (1.0f - p) * Z1;          // drop entries with cumsum <= 1-p
            for (int j = 0; j < m; ++j) {
                if (ve[j] > 0.f && (s_cs[j] > thr || j == 0)) { vkeep[sc++] = j; Z2 += ve[j]; }
            }
            const float logZ2 = __logf(Z2);
            int fill = 0;
            for (int t5 = 0; t5 < 5; ++t5) {            // top-5 logprobs + indices
                int oidx; float olp;
                if (t5 < sc) {
                    int j = vkeep[t5];
                    oidx = vti[j];
                    olp  = (vtv[j] - vmax) - logZ2;
                } else {                                 // fill: smallest unused vocab idx
                    for (;;) {
                        bool used = false;
                        for (int u2 = 0; u2 < sc && u2 < 5; ++u2)
                            if (vti[vkeep[u2]] == fill) { used = true; break; }
                        if (!used) break;
                        ++fill;
                    }
                    oidx = fill++; olp = NINF;
                }
                out[B + row * 5 + t5]          = (float)oidx;
                out[B + B * 5 + row * 5 + t5]  = olp;
            }
        } else if (lane == 0) {
            for (int t5 = 0; t5 < 5; ++t5) {
                out[B + row * 5 + t5] = 0.f;
                out[B + B * 5 + row * 5 + t5] = NINF;
            }
        }
        sc = __shfl(sc, 0, 32);
        Z2 = __shfl(Z2, 0, 32);

        // Gumbel-style sample: argmax over survivors of (p_j / q_j); sparse noise gather
        float rbest = NINF; int ibest = 0x7fffffff;
        for (int t2 = lane; t2 < sc; t2 += 32) {
            int j = vkeep[t2];
            int idx = vti[j];
            float u = noise[(size_t)row * V + idx];
            float q = -__logf(u);
            float r = (ve[j] / Z2) / q;
            if (r > rbest || (r == rbest && idx < ibest)) { rbest = r; ibest = idx; }
        }
        for (int off = 16; off > 0; off >>= 1) {
            float orv = __shfl_xor(rbest, off, 32);
            int   oid = __shfl_xor(ibest, off, 32);
            if (orv > rbest || (orv == rbest && oid < ibest)) { rbest = orv; ibest = oid; }
        }
        if (lane == 0) {
            int greedy  = (m > 0) ? vti[0] : 0;
            int sampled = (temp < SAMPLING_EPS || sc == 0) ? greedy : ibest;
            out[row] = (float)sampled;
        }
    }
}

extern "C" void kernel_launch(void* const* d_in, const int* in_sizes, int n_in,
                              void* d_out, int out_size, void* d_ws, size_t ws_size,
                              hipStream_t stream) {
    (void)n_in; (void)out_size; (void)d_ws; (void)ws_size;
    const float* logits      = (const float*)d_in[0];
    const float* temperature = (const float*)d_in[1];
    const int*   top_k       = (const int*)d_in[2];
    const float* top_p       = (const float*)d_in[3];
    const float* noise       = (const float*)d_in[4];
    const int B = in_sizes[1];             // 128 rows
    const int V = in_sizes[0] / B;         // 128000 vocab
    sampler_kernel<<<B, THREADS, 0, stream>>>(logits, temperature, top_k, top_p,
                                              noise, (float*)d_out, V, B);
}